// TransformerConv_6339371729524
// MI455X (gfx1250) — compile-verified
//
#include <hip/hip_runtime.h>
#include <hip/hip_bf16.h>
#include <math.h>

// Problem constants (match reference)
#define IN_C  128
#define EDGE_D 64
#define HEADS   4
#define OUT_C  32
#define HC    128   // HEADS*OUT_C

typedef __attribute__((ext_vector_type(2))) float v2f;
typedef __attribute__((ext_vector_type(8))) float v8f;

// Global-address-space pointer types: keep hot-loop loads/stores on the
// global_load/global_store path (LOADcnt/STOREcnt) instead of FLAT, which
// would also tie up DScnt and serialize against our LDS fragment loads.
typedef __attribute__((address_space(1))) const float g_cfloat;
typedef __attribute__((address_space(1))) float       g_float;

// ---------------------------------------------------------------------------
// CAS-based float atomic max (works for negative values too)
// ---------------------------------------------------------------------------
__device__ __forceinline__ void atomicMaxFloat(float* addr, float val) {
    unsigned int* ua = (unsigned int*)addr;
    unsigned int cur = __float_as_uint(*addr);
    while (__uint_as_float(cur) < val) {
        unsigned int assumed = cur;
        cur = atomicCAS(ua, assumed, __float_as_uint(val));
        if (cur == assumed) break;
    }
}

// ---------------------------------------------------------------------------
// Kernel 0: init denom[N*4] = 0, m[4] = -inf
// ---------------------------------------------------------------------------
__global__ void tc_init_kernel(float* __restrict__ denom, float* __restrict__ mbuf, int n) {
    int i = blockIdx.x * blockDim.x + threadIdx.x;
    if (i < n) denom[i] = 0.0f;
    if (i < HEADS) mbuf[i] = -INFINITY;
}

// ---------------------------------------------------------------------------
// Kernel 1: node GEMMs. One block = 16 node rows, 8 waves. mat = wave>>1 is
// wave-uniform, so each wave owns 4 column tiles (half a 128-col matrix) of
// exactly one of Q|K|V|S. f32 WMMA 16x16x4, K=128.
// S (= x@Ws^T + bs) is written straight into d_out as the residual base.
// ---------------------------------------------------------------------------
__global__ void __launch_bounds__(256) tc_node_gemm_kernel(
    const float* __restrict__ x,
    const float* __restrict__ Wq, const float* __restrict__ bq,
    const float* __restrict__ Wk, const float* __restrict__ bk,
    const float* __restrict__ Wv, const float* __restrict__ bv,
    const float* __restrict__ Ws, const float* __restrict__ bs,
    float* __restrict__ Q, float* __restrict__ K,
    float* __restrict__ V, float* __restrict__ out)
{
    __shared__ float x_s[16][IN_C];
    const int m0  = blockIdx.x * 16;
    const int tid = threadIdx.x;

    // Stage x tile [16 x 128] into LDS: 2048 floats / 256 threads = 8 each
    {
        int row  = tid >> 4;          // 0..15
        int col0 = (tid & 15) * 8;    // 0..120
        const float4* sp = (const float4*)(x + (size_t)(m0 + row) * IN_C + col0);
        float4 p0 = sp[0], p1 = sp[1];
        *(float4*)&x_s[row][col0]     = p0;
        *(float4*)&x_s[row][col0 + 4] = p1;
    }
    __syncthreads();

    const int wave = tid >> 5;
    const int lane = tid & 31;
    const int row  = lane & 15;   // M for A-frag, N for B-frag
    const int kh   = lane >> 4;   // K-half select
    const int mat  = wave >> 1;   // 0..3, uniform within the wave
    const int half = wave & 1;    // which 4 of the 8 column tiles

    // Wave-uniform pointer selection, pinned to the global address space.
    g_cfloat* W; g_cfloat* B; g_float* O;
    if      (mat == 0) { W = (g_cfloat*)Wq; B = (g_cfloat*)bq; O = (g_float*)Q;   }
    else if (mat == 1) { W = (g_cfloat*)Wk; B = (g_cfloat*)bk; O = (g_float*)K;   }
    else if (mat == 2) { W = (g_cfloat*)Wv; B = (g_cfloat*)bv; O = (g_float*)V;   }
    else               { W = (g_cfloat*)Ws; B = (g_cfloat*)bs; O = (g_float*)out; }

    #pragma unroll
    for (int t = 0; t < 4; ++t) {
        const int n0 = (half * 4 + t) * 16;
        v8f c = {};
        #pragma unroll 4
        for (int k0 = 0; k0 < IN_C; k0 += 4) {
            v2f a, b;
            a.x = x_s[row][k0 + 2 * kh];
            a.y = x_s[row][k0 + 2 * kh + 1];
            g_cfloat* wp = W + (size_t)(n0 + row) * IN_C + k0 + 2 * kh;
            b.x = wp[0];
            b.y = wp[1];
            c = __builtin_amdgcn_wmma_f32_16x16x4_f32(
                    false, a, false, b, (short)0, c, false, false);
        }
        const int col   = n0 + row;
        const float bia = B[col];
        #pragma unroll
        for (int r = 0; r < 8; ++r) {
            int rr = (lane < 16) ? r : (r + 8);
            O[(size_t)(m0 + rr) * HC + col] = c[r] + bia;
        }
    }
}

// ---------------------------------------------------------------------------
// Shared helper: compute e tile = edge_attr[e0:e0+16] @ We^T  ->  e_s[16][128]
// Block = 128 threads (4 waves); each wave does two 16x16 column tiles, K=64.
// ---------------------------------------------------------------------------
__device__ __forceinline__ void tc_compute_e_tile(
    const float* __restrict__ edge_attr, const float* __restrict__ We,
    int e0, float (*ea_s)[EDGE_D], float (*e_s)[HC])
{
    const int tid = threadIdx.x;   // 0..127
    // Stage edge_attr tile [16 x 64] into LDS: 1024 floats / 128 threads
    {
        int row  = tid >> 3;        // 0..15
        int col0 = (tid & 7) * 8;   // 0..56
        const float4* sp = (const float4*)(edge_attr + (size_t)(e0 + row) * EDGE_D + col0);
        float4 p0 = sp[0], p1 = sp[1];
        *(float4*)&ea_s[row][col0]     = p0;
        *(float4*)&ea_s[row][col0 + 4] = p1;
    }
    __syncthreads();

    g_cfloat* Wg = (g_cfloat*)We;
    const int wave = tid >> 5;
    const int lane = tid & 31;
    const int row  = lane & 15;
    const int kh   = lane >> 4;

    #pragma unroll
    for (int t = 0; t < 2; ++t) {
        const int n0 = (wave * 2 + t) * 16;
        v8f c = {};
        #pragma unroll 4
        for (int k0 = 0; k0 < EDGE_D; k0 += 4) {
            v2f a, b;
            a.x = ea_s[row][k0 + 2 * kh];
            a.y = ea_s[row][k0 + 2 * kh + 1];
            g_cfloat* wp = Wg + (size_t)(n0 + row) * EDGE_D + k0 + 2 * kh;
            b.x = wp[0];
            b.y = wp[1];
            c = __builtin_amdgcn_wmma_f32_16x16x4_f32(
                    false, a, false, b, (short)0, c, false, false);
        }
        const int col = n0 + row;
        #pragma unroll
        for (int r = 0; r < 8; ++r) {
            int rr = (lane < 16) ? r : (r + 8);
            e_s[rr][col] = c[r];
        }
    }
    __syncthreads();
}

// ---------------------------------------------------------------------------
// Kernel 2: per-edge e tile (WMMA) + raw attention logits + block/global max.
// alpha_raw[e][h] = <q_dst, k_src + e> / sqrt(OUT_C)
// ---------------------------------------------------------------------------
__global__ void __launch_bounds__(128) tc_edge_alpha_kernel(
    const float* __restrict__ edge_attr, const float* __restrict__ We,
    const int* __restrict__ ei,                     // [2*E]: src then dst
    const float* __restrict__ Q, const float* __restrict__ K,
    float* __restrict__ alpha, float* __restrict__ mbuf, int Etot)
{
    __shared__ float ea_s[16][EDGE_D];
    __shared__ float e_s[16][HC];
    __shared__ float ared[64];

    const int e0 = blockIdx.x * 16;
    tc_compute_e_tile(edge_attr, We, e0, ea_s, e_s);

    const int tid = threadIdx.x;
    if (tid < 64) {
        const int l = tid >> 2;    // edge in tile
        const int h = tid & 3;     // head
        const int eidx = e0 + l;
        const int s = ei[eidx];
        const int d = ei[Etot + eidx];
        g_cfloat* qp = (g_cfloat*)Q + (size_t)d * HC + h * OUT_C;
        g_cfloat* kp = (g_cfloat*)K + (size_t)s * HC + h * OUT_C;
        const float* ep = &e_s[l][h * OUT_C];
        float acc = 0.0f;
        #pragma unroll
        for (int c = 0; c < OUT_C; ++c)
            acc = fmaf(qp[c], kp[c] + ep[c], acc);
        acc *= 0.17677669529663687f;   // 1/sqrt(32)
        alpha[(size_t)eidx * HEADS + h] = acc;
        ared[tid] = acc;
    }
    __syncthreads();
    if (tid < HEADS) {
        float mx = ared[tid];
        #pragma unroll
        for (int l = 1; l < 16; ++l) mx = fmaxf(mx, ared[l * 4 + tid]);
        atomicMaxFloat(&mbuf[tid], mx);
    }
}

// ---------------------------------------------------------------------------
// Kernel 3: exp(alpha - m) in place + denom segment-sum via f32 atomics.
// ---------------------------------------------------------------------------
__global__ void tc_exp_denom_kernel(
    float* __restrict__ alpha, const int* __restrict__ ei,
    const float* __restrict__ mbuf, float* __restrict__ denom, int Etot)
{
    int idx = blockIdx.x * blockDim.x + threadIdx.x;
    if (idx >= Etot * HEADS) return;
    int eidx = idx >> 2;
    int h    = idx & 3;
    float ex = __expf(alpha[idx] - mbuf[h]);
    alpha[idx] = ex;
    int d = ei[Etot + eidx];
    atomicAdd(&denom[(size_t)d * HEADS + h], ex);
}

// ---------------------------------------------------------------------------
// Kernel 4: recompute e tile (WMMA), form msg = (V[src]+e)*alpha_norm,
// scatter-add into out[dst].
// ---------------------------------------------------------------------------
__global__ void __launch_bounds__(128) tc_scatter_kernel(
    const float* __restrict__ edge_attr, const float* __restrict__ We,
    const int* __restrict__ ei, const float* __restrict__ V,
    const float* __restrict__ alpha, const float* __restrict__ denom,
    float* __restrict__ out, int Etot)
{
    __shared__ float ea_s[16][EDGE_D];
    __shared__ float e_s[16][HC];

    const int e0 = blockIdx.x * 16;
    tc_compute_e_tile(edge_attr, We, e0, ea_s, e_s);

    const int tid = threadIdx.x;
    const int l   = tid >> 3;        // edge in tile (0..15)
    const int cg  = tid & 7;         // channel group
    const int c0  = cg * 16;         // 16 consecutive channels, one head
    const int h   = c0 >> 5;
    const int eidx = e0 + l;
    const int s = ei[eidx];
    const int d = ei[Etot + eidx];

    const float al = alpha[(size_t)eidx * HEADS + h] /
                     (denom[(size_t)d * HEADS + h] + 1e-16f);
    g_cfloat* vp = (g_cfloat*)V + (size_t)s * HC + c0;
    float*    op = out + (size_t)d * HC + c0;
    #pragma unroll
    for (int c = 0; c < 16; ++c) {
        float val = (vp[c] + e_s[l][c0 + c]) * al;
        atomicAdd(&op[c], val);
    }
}

// ---------------------------------------------------------------------------
extern "C" void kernel_launch(void* const* d_in, const int* in_sizes, int n_in,
                              void* d_out, int out_size, void* d_ws, size_t ws_size,
                              hipStream_t stream) {
    const float* x         = (const float*)d_in[0];
    const int*   ei        = (const int*)  d_in[1];   // [2, E]
    const float* edge_attr = (const float*)d_in[2];
    const float* Wq        = (const float*)d_in[3];
    const float* bq        = (const float*)d_in[4];
    const float* Wk        = (const float*)d_in[5];
    const float* bk        = (const float*)d_in[6];
    const float* Wv        = (const float*)d_in[7];
    const float* bv        = (const float*)d_in[8];
    const float* We        = (const float*)d_in[9];
    const float* Ws        = (const float*)d_in[10];
    const float* bs        = (const float*)d_in[11];
    float* out = (float*)d_out;

    const int N = in_sizes[0] / IN_C;     // 100000
    const int E = in_sizes[1] / 2;        // 500000

    // Workspace layout (floats): Q | K | V | alpha | denom | m
    float* Q     = (float*)d_ws;
    float* K     = Q     + (size_t)N * HC;
    float* V     = K     + (size_t)N * HC;
    float* alpha = V     + (size_t)N * HC;
    float* denom = alpha + (size_t)E * HEADS;
    float* mbuf  = denom + (size_t)N * HEADS;

    tc_init_kernel<<<(N * HEADS + 255) / 256, 256, 0, stream>>>(denom, mbuf, N * HEADS);

    tc_node_gemm_kernel<<<N / 16, 256, 0, stream>>>(
        x, Wq, bq, Wk, bk, Wv, bv, Ws, bs, Q, K, V, out);

    tc_edge_alpha_kernel<<<E / 16, 128, 0, stream>>>(
        edge_attr, We, ei, Q, K, alpha, mbuf, E);

    tc_exp_denom_kernel<<<(E * HEADS + 255) / 256, 256, 0, stream>>>(
        alpha, ei, mbuf, denom, E);

    tc_scatter_kernel<<<E / 16, 128, 0, stream>>>(
        edge_attr, We, ei, V, alpha, denom, out, E);
}